// TAGModel_71227737636876
// MI455X (gfx1250) — compile-verified
//
#include <hip/hip_runtime.h>
#include <hip/hip_bf16.h>
#include <stdint.h>

typedef __attribute__((ext_vector_type(2))) float v2f;
typedef __attribute__((ext_vector_type(8))) float v8f;

#define HC 256   // concat width = (K_HOPS+1)*64
#define D  64    // per-hop feature width

// ---------------- utility kernels ----------------

__global__ void fill_zero_kernel(float* __restrict__ p, long n) {
  long i = (long)blockIdx.x * blockDim.x + threadIdx.x;
  if (i < n) p[i] = 0.0f;
}

// zero the 64-wide column block (row stride 256) that a hop will accumulate into
__global__ void zero_strided_kernel(float* __restrict__ dst, long n64) {
  long i = (long)blockIdx.x * blockDim.x + threadIdx.x;
  if (i < n64) {
    long r = i >> 6; int c = (int)(i & 63);
    dst[r * HC + c] = 0.0f;
  }
}

// copy [N,64] contiguous -> column block 0 of [N,256]
__global__ void copy_strided_kernel(const float* __restrict__ src, float* __restrict__ dst, long n64) {
  long i = (long)blockIdx.x * blockDim.x + threadIdx.x;
  if (i < n64) {
    long r = i >> 6; int c = (int)(i & 63);
    dst[r * HC + c] = src[i];
  }
}

// ---------------- graph norm kernels ----------------

__global__ void deg_kernel(const int* __restrict__ dst, float* __restrict__ deg, int E) {
  int e = blockIdx.x * blockDim.x + threadIdx.x;
  if (e < E) atomicAdd(&deg[dst[e]], 1.0f);
}

__global__ void dinv_kernel(const float* __restrict__ deg, float* __restrict__ dinv, int n) {
  int i = blockIdx.x * blockDim.x + threadIdx.x;
  if (i < n) {
    float d = deg[i];
    dinv[i] = (d > 0.0f) ? rsqrtf(fmaxf(d, 1.0f)) : 0.0f;
  }
}

__global__ void norm_kernel(const int* __restrict__ src, const int* __restrict__ dst,
                            const float* __restrict__ dinv, float* __restrict__ norm, int E) {
  int e = blockIdx.x * blockDim.x + threadIdx.x;
  if (e < E) norm[e] = dinv[src[e]] * dinv[dst[e]];
}

// ---------------- sparse propagation: hout[dst] += norm * hin[src] ----------------
// 16 lanes per edge, each lane moves 4 contiguous floats (float4 gather + 4 f32 atomics).
// hin/hout are column blocks inside the [N,256] concat buffer (row stride HC).

__global__ void spmm_kernel(const int* __restrict__ src, const int* __restrict__ dst,
                            const float* __restrict__ norm,
                            const float* __restrict__ hin, float* __restrict__ hout, int E) {
  long t = (long)blockIdx.x * blockDim.x + threadIdx.x;
  int e = (int)(t >> 4);
  if (e >= E) return;
  int f = ((int)t & 15) << 2;
  int s = src[e], d = dst[e];
  float w = norm[e];
  const float4 v = *(const float4*)(hin + (long)s * HC + f);
  float* o = hout + (long)d * HC + f;
  atomicAdd(o + 0, w * v.x);
  atomicAdd(o + 1, w * v.y);
  atomicAdd(o + 2, w * v.z);
  atomicAdd(o + 3, w * v.w);
}

// ---------------- dense GEMM via V_WMMA_F32_16X16X4_F32, B staged in LDS ----------------
// Block = 8 waves = 8 consecutive row tiles; every wave computes ALL COLT column tiles
// (A fragment loaded once, reused COLT times). B panel [K x COLT*16] lives in LDS with
// K-pairs interleaved so each lane's B fragment is a single aligned ds_load_b64:
//   Bs[((k>>1)*COLT + ct)*32 + (n&15)*2 + (k&1)] = B[k][ct*16 + n]
// 32-bit WMMA layouts per ISA: A lane(0-15): M=lane, K={k0,k0+1}; lanes 16-31: K={k0+2,k0+3}.
// C/D: VGPR j -> M = j + 8*half, N = lane&15.

template <int COLT>
__global__ void gemm_wmma_kernel(const float* __restrict__ A, int lda,
                                 const float* __restrict__ B, int ncols,
                                 const float* __restrict__ bias,
                                 float* __restrict__ out, int ldo,
                                 int K, int row_tiles, int do_relu) {
  extern __shared__ float Bs[];  // K * COLT * 16 floats

  // cooperative fill of the zero-padded, pair-interleaved B panel
  const int tid = threadIdx.x;
  const int total = K * COLT * 16;
  for (int idx = tid; idx < total; idx += blockDim.x) {
    int row = idx / (COLT * 16);
    int col = idx - row * (COLT * 16);
    float v = (col < ncols) ? B[(long)row * ncols + col] : 0.0f;
    Bs[(((row >> 1) * COLT) + (col >> 4)) * 32 + ((col & 15) << 1) + (row & 1)] = v;
  }
  __syncthreads();

  const int wave = tid >> 5;
  const int rt = blockIdx.x * 8 + wave;
  if (rt >= row_tiles) return;               // wave-uniform: EXEC stays all-1s

  const int lane = tid & 31;
  const int half = lane >> 4;
  const int l    = lane & 15;

  const float* arow = A + (long)(rt * 16 + l) * lda;

  v8f acc[COLT];
#pragma unroll
  for (int c = 0; c < COLT; ++c) acc[c] = (v8f){};

  for (int k0 = 0; k0 < K; k0 += 4) {
    const int ka = k0 + (half << 1);
    v2f a;
    a.x = arow[ka];
    a.y = arow[ka + 1];
    const int k2 = (k0 >> 1) + half;
#pragma unroll
    for (int c = 0; c < COLT; ++c) {
      v2f b = *(const v2f*)&Bs[(k2 * COLT + c) * 32 + (l << 1)];
      acc[c] = __builtin_amdgcn_wmma_f32_16x16x4_f32(
          false, a, false, b, (short)0, acc[c], false, false);
    }
  }

#pragma unroll
  for (int c = 0; c < COLT; ++c) {
    int col = c * 16 + l;
    if (col >= ncols) continue;              // classifier tile 2: cols 40..47 masked
    float bv = bias[col];
#pragma unroll
    for (int j = 0; j < 8; ++j) {
      int row = rt * 16 + j + (half << 3);
      float v = acc[c][j] + bv;
      if (do_relu) v = v > 0.0f ? v : 0.0f;
      out[(long)row * ldo + col] = v;
    }
  }
}

// ---------------- launch ----------------

extern "C" void kernel_launch(void* const* d_in, const int* in_sizes, int n_in,
                              void* d_out, int out_size, void* d_ws, size_t ws_size,
                              hipStream_t stream) {
  const float* x  = (const float*)d_in[0];
  const int*   ei = (const int*)d_in[1];   // [2, E] (JAX default int32)
  const float* W1 = (const float*)d_in[2];
  const float* b1 = (const float*)d_in[3];
  const float* W2 = (const float*)d_in[4];
  const float* b2 = (const float*)d_in[5];
  const float* Wc = (const float*)d_in[6];
  const float* bc = (const float*)d_in[7];

  const int N = in_sizes[0] / D;       // 100000
  const int E = in_sizes[1] / 2;       // 1200000
  const int H = D;                     // 64
  const int NC = 40;

  const int* e_src = ei;
  const int* e_dst = ei + E;

  // workspace layout (floats)
  float* ws   = (float*)d_ws;
  float* deg  = ws;                         // N
  float* dinv = deg + N;                    // N
  float* nrm  = dinv + N;                   // E
  float* hcat = nrm + E;                    // N*256
  float* h    = hcat + (long)N * HC;        // N*64

  const int T = 256;
  const long n64 = (long)N * D;
  const int row_tiles = N / 16;             // 6250
  const int gemm_blocks = (row_tiles + 7) / 8;

  auto g1 = [&](long n) { return (int)((n + T - 1) / T); };

  // ---- gcn_norm (no self loops) ----
  fill_zero_kernel<<<g1(N), T, 0, stream>>>(deg, N);
  deg_kernel<<<g1(E), T, 0, stream>>>(e_dst, deg, E);
  dinv_kernel<<<g1(N), T, 0, stream>>>(deg, dinv, N);
  norm_kernel<<<g1(E), T, 0, stream>>>(e_src, e_dst, dinv, nrm, E);

  const long spmm_threads = (long)E * 16;

  // ---- TAGConv layer 1 ----
  copy_strided_kernel<<<g1(n64), T, 0, stream>>>(x, hcat, n64);
  for (int k = 0; k < 3; ++k) {
    zero_strided_kernel<<<g1(n64), T, 0, stream>>>(hcat + (k + 1) * D, n64);
    spmm_kernel<<<g1(spmm_threads), T, 0, stream>>>(e_src, e_dst, nrm,
                                                    hcat + k * D, hcat + (k + 1) * D, E);
  }
  gemm_wmma_kernel<4><<<gemm_blocks, T, HC * 64 * sizeof(float), stream>>>(
      hcat, HC, W1, H, b1, h, H, HC, row_tiles, /*relu=*/1);

  // ---- TAGConv layer 2 ----
  copy_strided_kernel<<<g1(n64), T, 0, stream>>>(h, hcat, n64);
  for (int k = 0; k < 3; ++k) {
    zero_strided_kernel<<<g1(n64), T, 0, stream>>>(hcat + (k + 1) * D, n64);
    spmm_kernel<<<g1(spmm_threads), T, 0, stream>>>(e_src, e_dst, nrm,
                                                    hcat + k * D, hcat + (k + 1) * D, E);
  }
  gemm_wmma_kernel<4><<<gemm_blocks, T, HC * 64 * sizeof(float), stream>>>(
      hcat, HC, W2, H, b2, h, H, HC, row_tiles, /*relu=*/1);

  // ---- classifier: [N,64] @ [64,40] + bc -> d_out (cols padded 40 -> 48) ----
  gemm_wmma_kernel<3><<<gemm_blocks, T, H * 48 * sizeof(float), stream>>>(
      h, H, Wc, NC, bc, (float*)d_out, NC, H, row_tiles, /*relu=*/0);
}